// DeBut_2dConv_65515431133409
// MI455X (gfx1250) — compile-verified
//
#include <hip/hip_runtime.h>
#include <hip/hip_bf16.h>

typedef float v2f __attribute__((ext_vector_type(2)));
typedef float v8f __attribute__((ext_vector_type(8)));

// ---------------------------------------------------------------------------
// Problem constants
// ---------------------------------------------------------------------------
#define HW      56
#define PIX     (HW * HW)          // 3136 pixels per image (divisible by 16)
#define CIN     128
#define COUT    256
#define BATCH   32
#define NPIX    (BATCH * PIX)      // 100352 total pixels
#define NTILES  16                 // 16 N-tiles of 16 output channels
#define KSTEPS  18                 // 72 features per N-tile / K=4 per WMMA
// Fragment-ordered weights: [nt][ks][lane][2] floats
#define WFRAG_PER_NT (KSTEPS * 32 * 2)        // 1152 floats
#define WFRAG_TOTAL  (NTILES * WFRAG_PER_NT)  // 18432 floats = 73728 bytes

// ---------------------------------------------------------------------------
// Kernel 1: collapse the three butterfly stages into dense 36x8 blocks,
// expanded to 16 block-diagonal 72x16 tiles, stored directly in the
// per-lane WMMA A-fragment order consumed by the main kernel.
//   A-frag (16x4 f32): lane l (l<16): M=l, K = ks*4+{0,1}
//                      lane l (>=16): M=l-16, K = ks*4+{2,3}
// ---------------------------------------------------------------------------
__global__ __launch_bounds__(256) void debut_build_w(
    const float* __restrict__ tw, float* __restrict__ wt)
{
    int idx = blockIdx.x * 256 + threadIdx.x;
    if (idx >= WFRAG_TOTAL) return;

    int j    = idx & 1;              // which of the 2 K values this lane holds
    int lane = (idx >> 1) & 31;
    int ks   = (idx >> 6) % KSTEPS;
    int nt   = idx / WFRAG_PER_NT;

    int k = ks * 4 + ((lane >= 16) ? 2 : 0) + j;   // 0..71 within N-tile
    int n = lane & 15;                             // output col 0..15

    float v = 0.0f;
    if ((k / 36) == (n / 8)) {                     // block-diagonal mask
        int blk = nt * 2 + (k / 36);               // n3 in 0..31
        int kd  = k % 36;                          // c3*9 + c2*3 + c1
        int od  = n & 7;                           // r3*4 + d3
        int c3 = kd / 9, c2 = (kd / 3) % 3, c1 = kd % 3;
        int r3 = od >> 2, d3 = od & 3;
        int d2 = d3 & 1,  r2 = d3 >> 1;
        // T3[n3,r3,c3,d3] = tw[3840 + n3*32 + d3*8 + r3*4 + c3]
        float t3 = tw[3840 + blk * 32 + d3 * 8 + r3 * 4 + c3];
        // T2[n2,r2,c2,d2] = tw[2304 + n2*12 + d2*6 + r2*3 + c2], n2 = 4*n3+c3
        float t2 = tw[2304 + (blk * 4 + c3) * 12 + d2 * 6 + r2 * 3 + c2];
        // T1[n1,r1,c1]    = tw[n1*6 + r1*3 + c1], n1 = 12*n3 + 3*c3 + c2, r1=d2
        float t1 = tw[(blk * 12 + c3 * 3 + c2) * 6 + d2 * 3 + c1];
        v = t1 * t2 * t3;
    }
    wt[idx] = v;
}

// ---------------------------------------------------------------------------
// Kernel 2: fused unfold + block-diagonal GEMM (transposed: Out[co][pix])
// via V_WMMA_F32_16X16X4_F32, + bias, written straight into NCHW output.
//   grid = (196, 16): x = pixel-tile groups, y = nt (output-channel tile)
//   block = 256 threads = 8 waves; each wave processes 4 pixel tiles of 16.
//   196 * 8 * 4 * 16 == 100352 == all pixels  -> no tails, EXEC always ~0.
// ---------------------------------------------------------------------------
__global__ __launch_bounds__(256) void debut_conv_wmma(
    const float* __restrict__ x, const float* __restrict__ wt,
    const float* __restrict__ bias, float* __restrict__ out)
{
    __shared__ float wlds[WFRAG_PER_NT];   // 4608 B: this block's N-tile weights

    const int nt = blockIdx.y;

    // Stage fragment-ordered weights for this nt into LDS (shared by 8 waves).
    const float* wsrc = wt + nt * WFRAG_PER_NT;
    for (int i = threadIdx.x; i < WFRAG_PER_NT; i += 256) wlds[i] = wsrc[i];
    __syncthreads();

    const int lane = threadIdx.x & 31;
    const int wave = threadIdx.x >> 5;
    const int hi   = lane >> 4;        // half-wave: 0 -> K {0,1}, 1 -> K {2,3}
    const int lp   = lane & 15;
    const int fsub = hi * 2;

    // Preload all 18 A fragments (2 f32 per lane per step) into registers.
    v2f afrag[KSTEPS];
#pragma unroll
    for (int ks = 0; ks < KSTEPS; ++ks)
        afrag[ks] = *(const v2f*)&wlds[(ks * 32 + lane) * 2];

    // Bias for the 8 accumulator rows this lane writes (co = nt*16 + hi*8 + j).
    float bv[8];
#pragma unroll
    for (int j = 0; j < 8; ++j) bv[j] = bias[nt * 16 + hi * 8 + j];

    const int fbase = nt * 72;

    for (int t = 0; t < 4; ++t) {
        const int tile = (blockIdx.x * 8 + wave) * 4 + t;
        const int p    = tile * 16 + lp;       // this lane's pixel (N index)
        const int b    = p / PIX;              // uniform across the wave
        const int pp   = p - b * PIX;
        const int ho   = pp / HW;
        const int wo   = pp - ho * HW;
        const float* xb = x + (size_t)b * (CIN * PIX);

        v8f acc = {0.f, 0.f, 0.f, 0.f, 0.f, 0.f, 0.f, 0.f};

#pragma unroll
        for (int ks = 0; ks < KSTEPS; ++ks) {
            // Features f0, f0+1 for this lane (same 3x3 tap, channels ch,ch+1).
            const int f0 = fbase + ks * 4 + fsub;
            const int kp = f0 >> 7;            // tap index 0..8
            const int ch = f0 & 127;
            const int dy = kp / 3 - 1;
            const int dx = kp - (kp / 3) * 3 - 1;
            const int iy = ho + dy;
            const int ix = wo + dx;
            const bool ok = (unsigned)iy < HW && (unsigned)ix < HW;
            const int base = ch * PIX + iy * HW + ix;

            v2f bfrag;
            bfrag.x = ok ? xb[base]       : 0.0f;   // K row f0,   pixel p
            bfrag.y = ok ? xb[base + PIX] : 0.0f;   // K row f0+1, pixel p

            // D(16x16 f32) = A(16x4) * B(4x16) + C
            acc = __builtin_amdgcn_wmma_f32_16x16x4_f32(
                false, afrag[ks], false, bfrag, (short)0, acc, false, false);
        }

        // acc[j] = Out[nt*16 + hi*8 + j][pixel p]; NCHW store, coalesced per
        // half-wave (16 contiguous floats along wo/pp).
        float* ob = out + (size_t)b * (COUT * PIX)
                        + (size_t)(nt * 16 + hi * 8) * PIX + pp;
#pragma unroll
        for (int j = 0; j < 8; ++j)
            ob[(size_t)j * PIX] = acc[j] + bv[j];
    }
}

// ---------------------------------------------------------------------------
extern "C" void kernel_launch(void* const* d_in, const int* in_sizes, int n_in,
                              void* d_out, int out_size, void* d_ws, size_t ws_size,
                              hipStream_t stream) {
    const float* x    = (const float*)d_in[0];   // (32,128,56,56) f32
    const float* tw   = (const float*)d_in[1];   // (4864,) f32
    const float* bias = (const float*)d_in[2];   // (256,) f32
    float* out = (float*)d_out;                  // (32,256,56,56) f32
    float* wt  = (float*)d_ws;                   // 73728 B scratch

    debut_build_w<<<dim3((WFRAG_TOTAL + 255) / 256), dim3(256), 0, stream>>>(tw, wt);

    dim3 grid(196, NTILES);   // 196*8 waves/nt * 4 tiles * 16 pix = 100352
    debut_conv_wmma<<<grid, dim3(256), 0, stream>>>(x, wt, bias, out);
}